// Decoder_14894946583396
// MI455X (gfx1250) — compile-verified
//
#include <hip/hip_runtime.h>

typedef __attribute__((ext_vector_type(2))) float v2f;
typedef __attribute__((ext_vector_type(8))) float v8f;

#define B_DIM 32
#define S_DIM 1024
#define R_DIM 1024
#define HID   512
#define OUT   1024

// ---------------------------------------------------------------------------
// GEMM1: h[32,512] = relu(rep[32,1024] @ W1^T + b1)
// One wave computes one 16x16 tile of h via V_WMMA_F32_16X16X4_F32,
// marching K in steps of 4. 2 M-tiles x 32 N-tiles = 64 waves = 8 blocks.
// ---------------------------------------------------------------------------
__global__ __launch_bounds__(256) void mlp_gemm1(const float* __restrict__ rep,
                                                 const float* __restrict__ W1,
                                                 const float* __restrict__ b1,
                                                 float* __restrict__ h)
{
    const int lane = threadIdx.x & 31;
    const int wave = threadIdx.x >> 5;
    const int gw   = blockIdx.x * 8 + wave;   // 0..63
    const int m0   = (gw & 1) * 16;           // batch tile: 0 or 16
    const int n0   = (gw >> 1) * 16;          // HID tile:  0..496
    const int half = lane >> 4;               // 0 / 1
    const int lr   = lane & 15;

    // A row for this lane (M = m0+lr), B row (N = n0+lr): W1 is [HID,R] row-major
    const float* arow = rep + (size_t)(m0 + lr) * R_DIM;
    const float* brow = W1  + (size_t)(n0 + lr) * R_DIM;

    v8f acc = {};
    #pragma unroll 8
    for (int k0 = 0; k0 < R_DIM; k0 += 4) {
        v2f a, b;
        // A 16x4 f32 layout: VGPR0 holds K = 2*half, VGPR1 holds K = 2*half+1
        a.x = arow[k0 + 2 * half + 0];
        a.y = arow[k0 + 2 * half + 1];
        // B 4x16 f32 layout mirrors A across N
        b.x = brow[k0 + 2 * half + 0];
        b.y = brow[k0 + 2 * half + 1];
        acc = __builtin_amdgcn_wmma_f32_16x16x4_f32(false, a, false, b,
                                                    (short)0, acc, false, false);
    }

    // C/D layout: VGPR v -> M = v + 8*half, N = lane&15
    const int n    = n0 + lr;
    const float bn = b1[n];
    #pragma unroll
    for (int v = 0; v < 8; ++v) {
        float val = acc[v] + bn;
        h[(size_t)(m0 + v + 8 * half) * HID + n] = fmaxf(val, 0.0f);
    }
}

// ---------------------------------------------------------------------------
// GEMM2: ovec[32,1024] = h[32,512] @ W2^T + b2
// 2 M-tiles x 64 N-tiles = 128 waves = 16 blocks.
// ---------------------------------------------------------------------------
__global__ __launch_bounds__(256) void mlp_gemm2(const float* __restrict__ h,
                                                 const float* __restrict__ W2,
                                                 const float* __restrict__ b2,
                                                 float* __restrict__ ovec)
{
    const int lane = threadIdx.x & 31;
    const int wave = threadIdx.x >> 5;
    const int gw   = blockIdx.x * 8 + wave;   // 0..127
    const int m0   = (gw & 1) * 16;           // batch tile
    const int n0   = (gw >> 1) * 16;          // OUT tile: 0..1008
    const int half = lane >> 4;
    const int lr   = lane & 15;

    const float* arow = h  + (size_t)(m0 + lr) * HID;
    const float* brow = W2 + (size_t)(n0 + lr) * HID;   // W2 is [OUT,HID] row-major

    v8f acc = {};
    #pragma unroll 8
    for (int k0 = 0; k0 < HID; k0 += 4) {
        v2f a, b;
        a.x = arow[k0 + 2 * half + 0];
        a.y = arow[k0 + 2 * half + 1];
        b.x = brow[k0 + 2 * half + 0];
        b.y = brow[k0 + 2 * half + 1];
        acc = __builtin_amdgcn_wmma_f32_16x16x4_f32(false, a, false, b,
                                                    (short)0, acc, false, false);
    }

    const int n    = n0 + lr;
    const float bn = b2[n];
    #pragma unroll
    for (int v = 0; v < 8; ++v) {
        ovec[(size_t)(m0 + v + 8 * half) * OUT + n] = acc[v] + bn;
    }
}

// ---------------------------------------------------------------------------
// Broadcast: out[b,s,:] = ovec[b,:] for all s.
// 134 MB of stores -> this is the roofline-dominant phase (~5.8 us at 23.3TB/s).
// Each thread holds one float4 of ovec in a register and streams 32 coalesced
// 16B stores (one per s in this block's chunk). Grid (S/32, B) = (32,32).
// ---------------------------------------------------------------------------
__global__ __launch_bounds__(256) void bcast_rows(const float* __restrict__ ovec,
                                                  float4* __restrict__ out)
{
    const int b = blockIdx.y;
    const int t = threadIdx.x;                       // 0..255 -> OUT/4 float4 slots
    const float4 v = ((const float4*)(ovec + (size_t)b * OUT))[t];
    const long long base =
        ((long long)b * S_DIM + (long long)blockIdx.x * 32) * (OUT / 4) + t;
    #pragma unroll
    for (int s = 0; s < 32; ++s) {
        out[base + (long long)s * (OUT / 4)] = v;
    }
}

extern "C" void kernel_launch(void* const* d_in, const int* in_sizes, int n_in,
                              void* d_out, int out_size, void* d_ws, size_t ws_size,
                              hipStream_t stream)
{
    const float* rep = (const float*)d_in[0];
    // d_in[1] = size_matrix: unused (reference only uses ones_like(size_matrix))
    const float* W1  = (const float*)d_in[2];
    const float* b1  = (const float*)d_in[3];
    const float* W2  = (const float*)d_in[4];
    const float* b2  = (const float*)d_in[5];
    float* out = (float*)d_out;

    // Workspace: h (32*512 f32 = 64KB) then ovec (32*1024 f32 = 128KB)
    float* h    = (float*)d_ws;
    float* ovec = h + (size_t)B_DIM * HID;

    mlp_gemm1<<<8, 256, 0, stream>>>(rep, W1, b1, h);
    mlp_gemm2<<<16, 256, 0, stream>>>(h, W2, b2, ovec);
    bcast_rows<<<dim3(S_DIM / 32, B_DIM), 256, 0, stream>>>(ovec, (float4*)out);
}